// GraphNorm_13632226197771
// MI455X (gfx1250) — compile-verified
//
#include <hip/hip_runtime.h>
#include <hip/hip_bf16.h>

typedef __attribute__((ext_vector_type(2))) float v2f;
typedef __attribute__((ext_vector_type(8))) float v8f;

#define NSEG 512
#define DF   128
#define SLICES 8

static __device__ __forceinline__ v8f wmma_acc(v2f a, v2f b, v8f c) {
    return __builtin_amdgcn_wmma_f32_16x16x4_f32(false, a, false, b, (short)0, c, false, false);
}

// ---------------- Kernel 0: segment offsets via binary search on sorted batch ----
__global__ void k_offsets(const int* __restrict__ batch, int N, int* __restrict__ offs) {
    int b = blockIdx.x * blockDim.x + threadIdx.x;
    if (b > NSEG) return;
    if (b == NSEG) { offs[NSEG] = N; return; }
    int lo = 0, hi = N;
    while (lo < hi) {
        int mid = (lo + hi) >> 1;
        if (batch[mid] < b) lo = mid + 1; else hi = mid;
    }
    offs[b] = lo;
}

// ---------------- Kernel 1: zero stat accumulators (graph-replay deterministic) --
__global__ void k_zero(float* __restrict__ p, int n) {
    int i = blockIdx.x * blockDim.x + threadIdx.x;
    if (i < n) p[i] = 0.0f;
}

// ---------------- Kernel 2: WMMA-based per-segment sum / sum-of-squares ----------
// grid = (NSEG, SLICES); block = 256 threads = 8 waves; wave w owns features
// [16w, 16w+16). V_WMMA_F32_16X16X4_F32 with all-ones A reduces 4 rows x 16
// features per issue. Main loop: 32 rows/iter, 16 unconditional b32 loads with
// immediate offsets (no branches, no per-load waits), 16 WMMAs. Guarded 4-row tail.
__global__ __launch_bounds__(256) void k_reduce(const float* __restrict__ x,
                                                const int* __restrict__ offs,
                                                float* __restrict__ segsum,
                                                float* __restrict__ segsq) {
    const int seg   = blockIdx.x;
    const int slice = blockIdx.y;
    const int s = offs[seg], e = offs[seg + 1];
    const int n = e - s;
    if (n <= 0) return;                      // uniform: EXEC stays full for WMMA
    const int per = (n + SLICES - 1) / SLICES;
    const int r0 = s + slice * per;
    const int r1 = min(r0 + per, e);
    if (r0 >= r1) return;                    // uniform

    const int lane = threadIdx.x & 31;
    const int wave = threadIdx.x >> 5;       // feature tile 0..7
    const int half = lane >> 4;              // 0/1
    const int col  = lane & 15;              // feature column within tile
    const int fbase = wave * 16 + col;

    v8f csum = {};
    v8f csq  = {};
    v2f ones; ones.x = 1.0f; ones.y = 1.0f;  // A = all-ones 16x4 (layout-proof)

    // per-lane base pointer; element 0 rows = r+half, element 1 rows = r+2+half
    const float* p = x + (size_t)(r0 + half) * DF + fbase;

    int r = r0;
    // -------- main body: 32 rows per iteration, fully unconditional ----------
    for (; r + 32 <= r1; r += 32, p += 32 * DF) {
        v2f b[8];
#pragma unroll
        for (int t = 0; t < 8; ++t) {
            b[t].x = p[t * 4 * DF];              // immediate offsets: t*2048 B
            b[t].y = p[t * 4 * DF + 2 * DF];     //                  + 1024 B
        }
#pragma unroll
        for (int t = 0; t < 8; ++t) {
            csum = wmma_acc(ones, b[t], csum);
            v2f bq; bq.x = b[t].x * b[t].x; bq.y = b[t].y * b[t].y;
            csq  = wmma_acc(ones, bq, csq);
        }
    }
    // -------- tail: 4-row steps with per-lane zero-fill ----------------------
    for (; r < r1; r += 4, p += 4 * DF) {
        v2f bm;
        bm.x = (r + half < r1)     ? p[0]      : 0.0f;
        bm.y = (r + 2 + half < r1) ? p[2 * DF] : 0.0f;
        csum = wmma_acc(ones, bm, csum);
        v2f bq; bq.x = bm.x * bm.x; bq.y = bm.y * bm.y;
        csq  = wmma_acc(ones, bq, csq);
    }

    if (lane < 16) {
        atomicAdd(&segsum[seg * DF + fbase], csum[0]);
        atomicAdd(&segsq [seg * DF + fbase], csq[0]);
    }
}

// ---------------- Kernel 3: fold stats into per-(seg,feature) affine P, Q --------
// out = centered*inv*scale + shift  ==  x*P[seg] + Q[seg]
//   norm^2 = sumsq - 2*a*m*sum + n*a^2*m^2
__global__ void k_finalize(const float* __restrict__ segsum, const float* __restrict__ segsq,
                           const int* __restrict__ offs,
                           const float* __restrict__ alpha, const float* __restrict__ scale,
                           const float* __restrict__ shift,
                           float* __restrict__ P, float* __restrict__ Q) {
    int idx = blockIdx.x * blockDim.x + threadIdx.x;
    if (idx >= NSEG * DF) return;
    int b = idx >> 7;
    int d = idx & (DF - 1);
    float cnt = (float)(offs[b + 1] - offs[b]);
    float ns  = fmaxf(cnt, 1.0f);
    float s1  = segsum[idx];
    float s2  = segsq[idx];
    float m   = s1 / ns;
    float a   = alpha[d];
    float norm2 = s2 - 2.0f * a * m * s1 + cnt * a * a * m * m;
    norm2 = fmaxf(norm2, 0.0f);
    float inv = floorf(sqrtf(ns)) / sqrtf(norm2);   // inf only for never-indexed empty segs
    float pv = inv * scale[d];
    P[idx] = pv;
    Q[idx] = shift[d] - a * m * pv;
}

// ---------------- Kernel 4: streaming apply, float4 vectorized -------------------
__global__ __launch_bounds__(256) void k_apply(const float4* __restrict__ x4,
                                               const int* __restrict__ batch,
                                               const float4* __restrict__ P4,
                                               const float4* __restrict__ Q4,
                                               float4* __restrict__ out4,
                                               long long total4) {
    long long idx = (long long)blockIdx.x * blockDim.x + threadIdx.x;
    if (idx >= total4) return;
    int row = (int)(idx >> 5);          // DF/4 = 32 float4 per row
    int g   = (int)(idx & 31);
    int seg = batch[row];
    float4 v = x4[idx];
    float4 p = P4[(size_t)seg * 32 + g];
    float4 q = Q4[(size_t)seg * 32 + g];
    float4 o;
    o.x = fmaf(v.x, p.x, q.x);
    o.y = fmaf(v.y, p.y, q.y);
    o.z = fmaf(v.z, p.z, q.z);
    o.w = fmaf(v.w, p.w, q.w);
    out4[idx] = o;
}

// ---------------- Launcher -------------------------------------------------------
extern "C" void kernel_launch(void* const* d_in, const int* in_sizes, int n_in,
                              void* d_out, int out_size, void* d_ws, size_t ws_size,
                              hipStream_t stream) {
    const float* x     = (const float*)d_in[0];
    const int*   batch = (const int*)d_in[1];
    const float* alpha = (const float*)d_in[2];
    const float* scale = (const float*)d_in[3];
    const float* shift = (const float*)d_in[4];
    float*       out   = (float*)d_out;

    const int N = in_sizes[0] / DF;

    // workspace layout
    char* ws = (char*)d_ws;
    int*   offs   = (int*)ws;                         ws += ((NSEG + 1) * sizeof(int) + 255) & ~255;
    float* segsum = (float*)ws;                       ws += NSEG * DF * sizeof(float);
    float* segsq  = (float*)ws;                       ws += NSEG * DF * sizeof(float);
    float* P      = (float*)ws;                       ws += NSEG * DF * sizeof(float);
    float* Q      = (float*)ws;

    k_offsets<<<(NSEG + 1 + 255) / 256, 256, 0, stream>>>(batch, N, offs);
    k_zero<<<(2 * NSEG * DF + 255) / 256, 256, 0, stream>>>(segsum, 2 * NSEG * DF); // segsq contiguous after segsum
    k_reduce<<<dim3(NSEG, SLICES), 256, 0, stream>>>(x, offs, segsum, segsq);
    k_finalize<<<(NSEG * DF + 255) / 256, 256, 0, stream>>>(segsum, segsq, offs,
                                                            alpha, scale, shift, P, Q);
    long long total4 = (long long)N * (DF / 4);
    k_apply<<<(unsigned)((total4 + 255) / 256), 256, 0, stream>>>((const float4*)x, batch,
                                                                  (const float4*)P, (const float4*)Q,
                                                                  (float4*)out, total4);
}